// PropagationCell_78451872628975
// MI455X (gfx1250) — compile-verified
//
#include <hip/hip_runtime.h>
#include <cmath>

// ---------------------------------------------------------------------------
// PropagationCell forward for MI455X (gfx1250, wave32).
// All dense GEMMs use V_WMMA_F32_16X16X4_F32 (full fp32 — workload is
// bandwidth-bound: ~134MB outputs vs ~10 GFLOP of matmul, so fp32 WMMA is the
// right precision/perf point). Inner loop is guard-free (K padded to 4) with
// b64 A-fragment loads, 4x unroll, and B-panel prefetch (global_prefetch_b8).
// Tiny-N linear heads (N=1,2,8) use VALU.
//
// Input order assumption: setup_inputs() dict insertion order; the nested
// params dict is flattened leaf-by-leaf in insertion order
// (_lin -> {w,b}, _bnp -> {g,b}) => d_in[11..60].
// ---------------------------------------------------------------------------

#define BSZ   128
#define NOBJ  16
#define IMGD  64
#define GSD   32
#define BNS   (BSZ * NOBJ)       // 2048
#define ZWH   64
#define NFD   64
#define NFHD  192
#define ADIMD 6
#define OBJD  74                  // 4+1+64+4+1
#define TIN   80                  // OBJD + ADIM
#define WBIN  136                 // 64+64+4+4
#define DEPIN 129                 // 64+64+1 (logical)
#define DEPPAD 132                // padded to multiple of 4

typedef float v2f __attribute__((ext_vector_type(2)));
typedef float v8f __attribute__((ext_vector_type(8)));

__device__ __forceinline__ float celu1(float x)     { return x > 0.f ? x : expm1f(x); }
__device__ __forceinline__ float softplus1(float x) { return x > 20.f ? x : log1pf(expf(x)); }
__device__ __forceinline__ float sigmoid1(float x)  { return 1.f / (1.f + expf(-x)); }

__device__ __forceinline__ float block_reduce_sum256(float v) {
  __shared__ float sh[256];
  int t = threadIdx.x;
  sh[t] = v; __syncthreads();
  for (int s = 128; s > 0; s >>= 1) { if (t < s) sh[t] += sh[t + s]; __syncthreads(); }
  float r = sh[0]; __syncthreads();
  return r;
}

// ------------------------------ GEMM (WMMA) --------------------------------
// C[M,N] = A[M,K] @ B[K,N] + bias[N]. One wave per 16x16 tile. K % 4 == 0,
// grid exactly tiles M,N: no guards anywhere in the hot loop.

struct PlainA {
  const float* A; int lda;
  // (r*lda + k) even and base 8B-aligned for all call sites -> single b64 load
  __device__ __forceinline__ v2f get2(int r, int k) const {
    return *(const v2f*)(A + (size_t)r * lda + k);
  }
};

// Fused loader for edge_enc: row = b*256 + i*16 + j over obj_rep[b*16+obj][74];
// cols [0,74) = receiver (i), [74,148) = sender (j). k is even, the region
// boundary (74) is even, row stride 296B is 8B-aligned -> b64 load is safe.
struct PairA {
  const float* obj_rep;
  __device__ __forceinline__ v2f get2(int r, int k) const {
    int b = r >> 8, i = (r >> 4) & 15, j = r & 15;
    const float* p = (k < OBJD)
        ? obj_rep + (size_t)(b * 16 + i) * OBJD + k
        : obj_rep + (size_t)(b * 16 + j) * OBJD + (k - OBJD);
    return *(const v2f*)p;
  }
};

template <typename AL>
__global__ __launch_bounds__(128)
void gemm_wmma_f32(AL al, const float* __restrict__ B, const float* __restrict__ bias,
                   float* __restrict__ C, int M, int N, int K) {
  const int lane = threadIdx.x;      // 0..31
  const int half = lane >> 4;        // 0 or 1
  const int l    = lane & 15;
  const int tm   = blockIdx.x * 16;
  const int tn   = (blockIdx.y * blockDim.y + threadIdx.y) * 16;

  const float* Bp = B + tn + l;      // this lane's B column
  const int    row = tm + l;         // this lane's A row

  v8f acc = {};
  // A 16x4 f32 frag: lanes 0-15 -> K = k,k+1 ; lanes 16-31 -> K = k+2,k+3.
  // B 4x16 f32 frag: mirrored K split; N striped across 16 lanes.
  #pragma unroll 4
  for (int k = 0; k < K; k += 4) {
    const int ka = k + 2 * half;
    v2f a = al.get2(row, ka);
    v2f b;
    b.x = Bp[(size_t)ka * N];
    b.y = Bp[(size_t)(ka + 1) * N];
    // speculative prefetch of the B panel 4 steps ahead (global_prefetch_b8)
    __builtin_prefetch(Bp + (size_t)(ka + 16) * N, 0, 1);
    acc = __builtin_amdgcn_wmma_f32_16x16x4_f32(false, a, false, b, (short)0, acc,
                                                false, false);
  }
  const float bv = bias[tn + l];
  #pragma unroll
  for (int r = 0; r < 8; ++r) {
    int orow = tm + r + 8 * half;    // C/D layout: VGPR r -> M=r (lanes 0-15), M=r+8
    C[(size_t)orow * N + tn + l] = acc[r] + bv;
  }
}

// ------------------------------ small linear -------------------------------
__global__ __launch_bounds__(256)
void linear_small(const float* __restrict__ A, const float* __restrict__ W,
                  const float* __restrict__ bias, float* __restrict__ C,
                  int M, int N, int K) {
  int t = blockIdx.x * 256 + threadIdx.x;
  if (t >= M * N) return;
  int r = t / N, n = t - r * N;
  float s = bias[n];
  const float* a = A + (size_t)r * K;
  for (int k = 0; k < K; ++k) s += a[k] * W[k * N + n];
  C[t] = s;
}

// zero-pad a KxN weight to KpadxN (for dep1: 129 -> 132 rows)
__global__ __launch_bounds__(256)
void k_pad_w(const float* __restrict__ W, float* __restrict__ Wp,
             int Kin, int Kpad, int N) {
  int t = blockIdx.x * 256 + threadIdx.x;
  if (t >= Kpad * N) return;
  int k = t / N, n = t - k * N;
  Wp[t] = (k < Kin) ? W[k * N + n] : 0.f;
}

// ------------------------------ batchnorm ----------------------------------
__global__ __launch_bounds__(256)
void bn_stats_celu(const float* __restrict__ Y, float* __restrict__ mean,
                   float* __restrict__ rstd, int M, int N) {
  int col = blockIdx.x;
  float s = 0.f, ss = 0.f;
  for (int r = threadIdx.x; r < M; r += 256) {
    float v = celu1(Y[(size_t)r * N + col]);
    s += v; ss += v * v;
  }
  s  = block_reduce_sum256(s);
  ss = block_reduce_sum256(ss);
  if (threadIdx.x == 0) {
    float m = s / M;
    float var = ss / M - m * m;
    mean[col] = m;
    rstd[col] = rsqrtf(var + 1e-5f);
  }
}

__global__ __launch_bounds__(256)
void bn_apply_celu(float* __restrict__ Y, const float* __restrict__ mean,
                   const float* __restrict__ rstd, const float* __restrict__ g,
                   const float* __restrict__ b, int total, int N) {
  int t = blockIdx.x * 256 + threadIdx.x;
  if (t >= total) return;
  int col = t % N;
  float v = celu1(Y[t]);
  Y[t] = (v - mean[col]) * rstd[col] * g[col] + b[col];
}

__global__ __launch_bounds__(256)
void k_celu_inplace(float* __restrict__ Y, int total) {
  int t = blockIdx.x * 256 + threadIdx.x;
  if (t >= total) return;
  Y[t] = celu1(Y[t]);
}

// ------------------------------ concat builders ----------------------------
__global__ __launch_bounds__(256)
void k_obj_rep(const float* zw, const float* zp, const float* zwh,
               const float* zwb, const float* zd, float* out) {
  int t = blockIdx.x * 256 + threadIdx.x;
  if (t >= BNS * OBJD) return;
  int o = t / OBJD, c = t - o * OBJD;
  float v;
  if      (c < 4)  v = zw[o * 4 + c];
  else if (c < 5)  v = zp[o];
  else if (c < 69) v = zwh[o * 64 + (c - 5)];
  else if (c < 73) v = zwb[o * 4 + (c - 69)];
  else             v = zd[o];
  out[t] = v;
}

__global__ __launch_bounds__(256)
void k_objact(const float* objrep, const float* act, const float* nt, float* out) {
  int t = blockIdx.x * 256 + threadIdx.x;
  if (t >= BNS * TIN) return;
  int o = t / TIN, c = t - o * TIN;
  if (c < OBJD) out[t] = objrep[(size_t)o * OBJD + c];
  else {
    int b = o >> 4, i = o & 15;
    out[t] = act[b * ADIMD + (c - OBJD)] * nt[b * 16 + i];
  }
}

__global__ __launch_bounds__(256)
void k_wbinp(const float* transit, const float* zwh, const float* zw,
             const float* zwb, float* out) {
  int t = blockIdx.x * 256 + threadIdx.x;
  if (t >= BNS * WBIN) return;
  int o = t / WBIN, c = t - o * WBIN;
  float v;
  if      (c < 64)  v = transit[o * 64 + c];
  else if (c < 128) v = zwh[o * 64 + (c - 64)];
  else if (c < 132) v = zw[o * 4 + (c - 128)];
  else              v = zwb[o * 4 + (c - 132)];
  out[t] = v;
}

// builds the zero-padded (132-col) depth input
__global__ __launch_bounds__(256)
void k_depinp(const float* transit, const float* zwhat, const float* zd, float* out) {
  int t = blockIdx.x * 256 + threadIdx.x;
  if (t >= BNS * DEPPAD) return;
  int o = t / DEPPAD, c = t - o * DEPPAD;
  float v;
  if      (c < 64)  v = transit[o * 64 + c];
  else if (c < 128) v = zwhat[o * 64 + (c - 64)];
  else if (c < 129) v = zd[o];
  else              v = 0.f;
  out[t] = v;
}

// ------------------------------ node gumbel-softmax ------------------------
__global__ __launch_bounds__(128)
void k_node_type(const float* __restrict__ logits, const float* __restrict__ gn,
                 float* __restrict__ nt) {
  int b = blockIdx.x * blockDim.x + threadIdx.x;
  if (b >= BSZ) return;
  float v[16]; float mx = -1e30f;
  for (int i = 0; i < 16; ++i) {
    float g = -logf(-logf(gn[b * 16 + i] + 1e-20f) + 1e-20f);
    v[i] = logits[b * 16 + i] + g;
    mx = fmaxf(mx, v[i]);
  }
  float s = 0.f;
  for (int i = 0; i < 16; ++i) { v[i] = expf(v[i] - mx); s += v[i]; }
  float inv = 1.f / s;
  for (int i = 0; i < 16; ++i) nt[b * 16 + i] = v[i] * inv;
}

// ------------------------------ z_where / z_what / z_depth -----------------
__global__ __launch_bounds__(256)
void k_zwhere(const float* __restrict__ wb, const float* __restrict__ eps,
              const float* __restrict__ zwpre, float* __restrict__ zw) {
  int o = blockIdx.x * 256 + threadIdx.x;
  if (o >= BNS) return;
  float zwb[4];
  #pragma unroll
  for (int k = 0; k < 4; ++k) {
    float m = wb[o * 8 + k];
    float s = softplus1(wb[o * 8 + 4 + k] - 2.0f);   // Z_WHERE_STD_BIAS
    zwb[k] = m + s * eps[o * 4 + k];
  }
  float shx = zwpre[o * 4 + 2] + 0.2f * tanhf(zwb[2]);
  float shy = zwpre[o * 4 + 3] + 0.2f * tanhf(zwb[3]);
  float sr0 = tanhf(zwb[0]), sr1 = tanhf(zwb[1]);
  float scale = 0.25f + 0.1f * sr0;     // SIZE_ANC + VAR_S * sr
  float ratio = 1.0f + 0.5f * sr1;      // RATIO_ANC + VAR_ANC * sr
  float rs = sqrtf(ratio);
  zw[o * 4 + 0] = scale / rs;
  zw[o * 4 + 1] = scale * rs;
  zw[o * 4 + 2] = fminf(fmaxf(shx, -1.05f), 1.05f);
  zw[o * 4 + 3] = fminf(fmaxf(shy, -1.05f), 1.05f);
}

__global__ __launch_bounds__(256)
void k_zwhat(const float* __restrict__ wt, const float* __restrict__ eps,
             float* __restrict__ zwhat) {
  int t = blockIdx.x * 256 + threadIdx.x;
  if (t >= BNS * ZWH) return;
  int o = t >> 6, d = t & 63;
  zwhat[t] = wt[o * 128 + d] + softplus1(wt[o * 128 + 64 + d]) * eps[t];
}

__global__ __launch_bounds__(256)
void k_zdepth(const float* __restrict__ dep, const float* __restrict__ eps,
              float* __restrict__ zd) {
  int o = blockIdx.x * 256 + threadIdx.x;
  if (o >= BNS) return;
  zd[o] = dep[o * 2] + softplus1(dep[o * 2 + 1]) * eps[o];
}

// ------------------------------ spatial transforms -------------------------
__device__ __forceinline__ float tap2(const float* img, int W, int H, int xi, int yi) {
  if (xi < 0 || xi >= W || yi < 0 || yi >= H) return 0.f;
  return img[yi * W + xi];
}
__device__ __forceinline__ float bilin(const float* img, int W, int H, float xs, float ys) {
  float x0f = floorf(xs), y0f = floorf(ys);
  int x0 = (int)x0f, y0 = (int)y0f;
  float wx = xs - x0f, wy = ys - y0f;
  float v00 = tap2(img, W, H, x0,     y0);
  float v10 = tap2(img, W, H, x0 + 1, y0);
  float v01 = tap2(img, W, H, x0,     y0 + 1);
  float v11 = tap2(img, W, H, x0 + 1, y0 + 1);
  return v00 * (1.f - wx) * (1.f - wy) + v10 * wx * (1.f - wy)
       + v01 * (1.f - wx) * wy        + v11 * wx * wy;
}

__global__ __launch_bounds__(256)
void k_xatt(const float* __restrict__ x, const float* __restrict__ zw,
            float* __restrict__ xatt) {
  int t = blockIdx.x * 256 + threadIdx.x;
  if (t >= BNS * 3 * GSD * GSD) return;
  int o   = t / (3 * GSD * GSD);
  int rem = t - o * (3 * GSD * GSD);
  int c   = rem / (GSD * GSD);
  int p   = rem - c * (GSD * GSD);
  int px = p & (GSD - 1), py = p >> 5;
  int b = o >> 4;
  float sx = zw[o * 4 + 0], sy = zw[o * 4 + 1];
  float tx = zw[o * 4 + 2], ty = zw[o * 4 + 3];
  float gx = ((px + 0.5f) / GSD) * 2.f - 1.f;
  float gy = ((py + 0.5f) / GSD) * 2.f - 1.f;
  float u = sx * gx + tx, v = sy * gy + ty;
  float xs = ((u + 1.f) * IMGD - 1.f) * 0.5f;
  float ys = ((v + 1.f) * IMGD - 1.f) * 0.5f;
  xatt[t] = bilin(x + (size_t)(b * 3 + c) * IMGD * IMGD, IMGD, IMGD, xs, ys);
}

__global__ __launch_bounds__(256)
void k_prepatt(const float* __restrict__ dec, const float* __restrict__ zdepth,
               float* __restrict__ yatt, float* __restrict__ imp) {
  int t = blockIdx.x * 256 + threadIdx.x;
  if (t >= BNS * GSD * GSD) return;
  int o = t / (GSD * GSD), p = t - o * (GSD * GSD);
  const float* d = dec + (size_t)o * 4 * GSD * GSD;
  float al = sigmoid1(d[3 * GSD * GSD + p]);
  float* yo = yatt + (size_t)o * 3 * GSD * GSD;
  yo[p]                 = al * sigmoid1(d[p]);
  yo[GSD * GSD + p]     = al * sigmoid1(d[GSD * GSD + p]);
  yo[2 * GSD * GSD + p] = al * sigmoid1(d[2 * GSD * GSD + p]);
  imp[t] = al * sigmoid1(-zdepth[o]);
}

__global__ __launch_bounds__(256)
void k_inverse(const float* __restrict__ yatt, const float* __restrict__ imp,
               const float* __restrict__ zw, float* __restrict__ oy,
               float* __restrict__ oi) {
  int t = blockIdx.x * 256 + threadIdx.x;
  if (t >= BNS * IMGD * IMGD) return;
  int o = t / (IMGD * IMGD), p = t - o * (IMGD * IMGD);
  int px = p & (IMGD - 1), py = p >> 6;
  float sx = zw[o * 4 + 0], sy = zw[o * 4 + 1];
  float tx = zw[o * 4 + 2], ty = zw[o * 4 + 3];
  float isx = 1.f / sx, isy = 1.f / sy;
  float itx = -tx * isx, ity = -ty * isy;
  float gx = ((px + 0.5f) / IMGD) * 2.f - 1.f;
  float gy = ((py + 0.5f) / IMGD) * 2.f - 1.f;
  float u = isx * gx + itx, v = isy * gy + ity;
  float xs = ((u + 1.f) * GSD - 1.f) * 0.5f;
  float ys = ((v + 1.f) * GSD - 1.f) * 0.5f;
  const float* ya = yatt + (size_t)o * 3 * GSD * GSD;
  oy[(size_t)(o * 3 + 0) * IMGD * IMGD + p] = bilin(ya,                 GSD, GSD, xs, ys);
  oy[(size_t)(o * 3 + 1) * IMGD * IMGD + p] = bilin(ya + GSD * GSD,     GSD, GSD, xs, ys);
  oy[(size_t)(o * 3 + 2) * IMGD * IMGD + p] = bilin(ya + 2 * GSD * GSD, GSD, GSD, xs, ys);
  oi[(size_t)o * IMGD * IMGD + p]           = bilin(imp + (size_t)o * GSD * GSD, GSD, GSD, xs, ys);
}

// ------------------------------ KL reductions ------------------------------
__global__ __launch_bounds__(256)
void k_kl_what(const float* __restrict__ enc, const float* __restrict__ wt,
               const float* __restrict__ zpres, float* __restrict__ out) {
  int b = blockIdx.x;
  float acc = 0.f;
  for (int idx = threadIdx.x; idx < NOBJ * ZWH; idx += 256) {
    int i = idx >> 6, d = idx & 63;
    int go = b * 16 + i;
    float tm = wt[go * 128 + d],  ts = softplus1(wt[go * 128 + 64 + d]);
    float em = enc[go * 128 + d], es = softplus1(enc[go * 128 + 64 + d]);
    float diff = tm - em;
    float kl = logf(es / ts) + (ts * ts + diff * diff) / (2.f * es * es) - 0.5f;
    float mask = (zpres[go] != 0.f) ? 1.f : 0.f;
    acc += kl * mask;
  }
  acc = block_reduce_sum256(acc);
  if (threadIdx.x == 0) out[b] = acc;
}

__global__ __launch_bounds__(256)
void k_kl_edge(const float* __restrict__ el, const float* __restrict__ zpres,
               float* __restrict__ out) {
  const float L0 = -0.2231435513f;  // log(0.8)
  const float L1 = -1.6094379124f;  // log(0.2)
  int b = blockIdx.x;
  int t = threadIdx.x;              // pair index, exactly 256 pairs
  int i = t >> 4, j = t & 15;
  size_t ij = (size_t)((b * 16 + i) * 16 + j) * 2;
  size_t ji = (size_t)((b * 16 + j) * 16 + i) * 2;
  float e0 = 0.5f * (el[ij]     + el[ji]);
  float e1 = 0.5f * (el[ij + 1] + el[ji + 1]);
  float m = fmaxf(e0, e1);
  float x0 = expf(e0 - m), x1 = expf(e1 - m);
  float s = x0 + x1, ls = logf(s);
  float q0 = x0 / s, q1 = x1 / s;
  float kl = q0 * ((e0 - m - ls) - L0) + q1 * ((e1 - m - ls) - L1);
  float mask = ((zpres[b * 16 + i] != 0.f) ? 1.f : 0.f)
             * ((zpres[b * 16 + j] != 0.f) ? 1.f : 0.f);
  float acc = block_reduce_sum256(kl * mask);
  if (threadIdx.x == 0) out[b] = acc;
}

// ===========================================================================
extern "C" void kernel_launch(void* const* d_in, const int* in_sizes, int n_in,
                              void* d_out, int out_size, void* d_ws, size_t ws_size,
                              hipStream_t stream) {
  (void)in_sizes; (void)n_in; (void)out_size; (void)ws_size;

  const float* x           = (const float*)d_in[0];
  const float* act         = (const float*)d_in[1];
  const float* z_what_pre  = (const float*)d_in[2];
  const float* z_where_pre = (const float*)d_in[3];
  const float* z_wb_pre    = (const float*)d_in[4];
  const float* z_depth_pre = (const float*)d_in[5];
  const float* z_pres_pre  = (const float*)d_in[6];
  const float* g_node      = (const float*)d_in[7];
  const float* eps_where   = (const float*)d_in[8];
  const float* eps_what    = (const float*)d_in[9];
  const float* eps_depth   = (const float*)d_in[10];

  auto P = [&](int i) { return (const float*)d_in[i]; };
  const float *Wnode = P(11), *bnode = P(12), *gnbn = P(13), *bnbn = P(14);
  const float *Wnp   = P(15), *bnp_  = P(16);
  const float *Wee   = P(17), *bee   = P(18), *gebn = P(19), *bebn = P(20);
  const float *Wep   = P(21), *bep   = P(22);
  const float *Wtm1  = P(23), *btm1  = P(24), *gtb1 = P(25), *btb1 = P(26);
  const float *Wtm2  = P(27), *btm2  = P(28), *gtb2 = P(29), *btb2 = P(30);
  const float *Wtm3  = P(31), *btm3  = P(32), *gtb3 = P(33), *btb3 = P(34);
  const float *Wtm4  = P(35), *btm4  = P(36);
  const float *Wwb1  = P(37), *bwb1  = P(38), *Wwb2 = P(39), *bwb2 = P(40);
  const float *Wwt1  = P(41), *bwt1  = P(42), *gwtb = P(43), *bwtb = P(44);
  const float *Wwt2  = P(45), *bwt2  = P(46);
  const float *Wd1   = P(47), *bd1   = P(48), *gdb  = P(49), *bdb  = P(50);
  const float *Wd2   = P(51), *bd2   = P(52);
  const float *We1   = P(53), *be1   = P(54), *We2  = P(55), *be2  = P(56);
  const float *Wdc1  = P(57), *bdc1  = P(58), *Wdc2 = P(59), *bdc2 = P(60);

  // ----- workspace carve (floats) -----
  float* ws = (float*)d_ws;
  size_t off = 0;
  auto alloc = [&](size_t n) { float* p = ws + off; off += n; return p; };

  float* obj_rep     = alloc((size_t)BNS * OBJD);
  float* node_y      = alloc((size_t)BNS * NFHD);
  float* stat_mean   = alloc(256);
  float* stat_rstd   = alloc(256);
  float* node_logits = alloc(BNS);
  float* node_type   = alloc(BNS);
  float* edge_y      = alloc((size_t)BSZ * 256 * NFHD);   // 32768 x 192
  float* edge_logits = alloc((size_t)BSZ * 256 * 2);
  float* objact      = alloc((size_t)BNS * TIN);
  float* tbufA       = alloc((size_t)BNS * NFD);
  float* tbufB       = alloc((size_t)BNS * NFD);
  float* transit     = alloc((size_t)BNS * NFD);
  float* wb_inp      = alloc((size_t)BNS * WBIN);
  float* wb_h        = alloc((size_t)BNS * NFD);
  float* wb_out      = alloc((size_t)BNS * 8);
  float* z_where     = alloc((size_t)BNS * 4);
  float* x_att       = alloc((size_t)BNS * 3 * GSD * GSD); // reused as y_att later
  float* enc_h       = alloc((size_t)BNS * 256);
  float* enc_out     = alloc((size_t)BNS * 128);
  float* wt_out      = alloc((size_t)BNS * 128);
  float* z_what      = alloc((size_t)BNS * ZWH);
  float* dep_inp     = alloc((size_t)BNS * DEPPAD);
  float* Wd1p        = alloc((size_t)DEPPAD * NFD);        // zero-padded dep1.w
  float* dep_out     = alloc((size_t)BNS * 2);
  float* z_depth     = alloc(BNS);
  float* dec_h       = alloc((size_t)BNS * 256);
  float* dec_out     = alloc((size_t)BNS * 4096);
  float* imp         = alloc((size_t)BNS * GSD * GSD);
  float* wt_y  = tbufB;   // dead after tm3 input consumed
  float* dep_y = tbufA;   // dead after tm4 input consumed
  float* y_att = x_att;   // x_att dead after enc1

  // ----- output layout -----
  float* out_y   = (float*)d_out;                             // 2048*3*64*64
  float* out_imp = out_y + (size_t)BNS * 3 * IMGD * IMGD;     // 2048*64*64
  float* out_klw = out_imp + (size_t)BNS * IMGD * IMGD;       // 128
  float* out_kle = out_klw + BSZ;                             // 128

  auto EWB = [](size_t n) { return dim3((unsigned)((n + 255) / 256)); };
  auto gemmP = [&](const float* A, int lda, const float* W, const float* bias,
                   float* C, int M, int N, int K) {
    dim3 g((unsigned)(M / 16), (unsigned)((N + 63) / 64));
    gemm_wmma_f32<PlainA><<<g, dim3(32, 4), 0, stream>>>(PlainA{A, lda}, W, bias,
                                                         C, M, N, K);
  };
  auto bnrun = [&](float* Y, const float* g, const float* b, int M, int N) {
    bn_stats_celu<<<N, 256, 0, stream>>>(Y, stat_mean, stat_rstd, M, N);
    size_t tot = (size_t)M * N;
    bn_apply_celu<<<EWB(tot), 256, 0, stream>>>(Y, stat_mean, stat_rstd, g, b,
                                                (int)tot, N);
  };

  // 1. obj_rep
  k_obj_rep<<<EWB((size_t)BNS * OBJD), 256, 0, stream>>>(
      z_where_pre, z_pres_pre, z_what_pre, z_wb_pre, z_depth_pre, obj_rep);

  // 2-4. node path
  gemmP(z_what_pre, ZWH, Wnode, bnode, node_y, BNS, NFHD, ZWH);
  bnrun(node_y, gnbn, bnbn, BNS, NFHD);
  linear_small<<<EWB(BNS), 256, 0, stream>>>(node_y, Wnp, bnp_, node_logits,
                                             BNS, 1, NFHD);
  // 5. gumbel softmax -> node_type
  k_node_type<<<1, 128, 0, stream>>>(node_logits, g_node, node_type);

  // 6-8. edge path (fused pair construction, K = 148 % 4 == 0)
  {
    dim3 g((unsigned)(BSZ * 256 / 16), (unsigned)(NFHD / 64));
    gemm_wmma_f32<PairA><<<g, dim3(32, 4), 0, stream>>>(
        PairA{obj_rep}, Wee, bee, edge_y, BSZ * 256, NFHD, 2 * OBJD);
  }
  bnrun(edge_y, gebn, bebn, BSZ * 256, NFHD);
  linear_small<<<EWB((size_t)BSZ * 256 * 2), 256, 0, stream>>>(
      edge_y, Wep, bep, edge_logits, BSZ * 256, 2, NFHD);
  // 9. kl_edge_type -> d_out
  k_kl_edge<<<BSZ, 256, 0, stream>>>(edge_logits, z_pres_pre, out_kle);

  // 10-14. transit MLP
  k_objact<<<EWB((size_t)BNS * TIN), 256, 0, stream>>>(obj_rep, act, node_type, objact);
  gemmP(objact, TIN, Wtm1, btm1, tbufA, BNS, NFD, TIN);
  bnrun(tbufA, gtb1, btb1, BNS, NFD);
  gemmP(tbufA, NFD, Wtm2, btm2, tbufB, BNS, NFD, NFD);
  bnrun(tbufB, gtb2, btb2, BNS, NFD);
  gemmP(tbufB, NFD, Wtm3, btm3, tbufA, BNS, NFD, NFD);
  bnrun(tbufA, gtb3, btb3, BNS, NFD);
  gemmP(tbufA, NFD, Wtm4, btm4, transit, BNS, NFD, NFD);

  // 15-18. where-bias head -> z_where
  k_wbinp<<<EWB((size_t)BNS * WBIN), 256, 0, stream>>>(transit, z_what_pre,
                                                       z_where_pre, z_wb_pre, wb_inp);
  gemmP(wb_inp, WBIN, Wwb1, bwb1, wb_h, BNS, NFD, WBIN);
  k_celu_inplace<<<EWB((size_t)BNS * NFD), 256, 0, stream>>>(wb_h, BNS * NFD);
  linear_small<<<EWB((size_t)BNS * 8), 256, 0, stream>>>(wb_h, Wwb2, bwb2, wb_out,
                                                         BNS, 8, NFD);
  k_zwhere<<<EWB(BNS), 256, 0, stream>>>(wb_out, eps_where, z_where_pre, z_where);

  // 19. forward spatial transform (crop)
  k_xatt<<<EWB((size_t)BNS * 3 * GSD * GSD), 256, 0, stream>>>(x, z_where, x_att);

  // 20-21. encoder
  gemmP(x_att, 3 * GSD * GSD, We1, be1, enc_h, BNS, 256, 3 * GSD * GSD);
  k_celu_inplace<<<EWB((size_t)BNS * 256), 256, 0, stream>>>(enc_h, BNS * 256);
  gemmP(enc_h, 256, We2, be2, enc_out, BNS, 128, 256);

  // 22-24. what-transit head -> z_what
  gemmP(transit, NFD, Wwt1, bwt1, wt_y, BNS, NFD, NFD);
  bnrun(wt_y, gwtb, bwtb, BNS, NFD);
  gemmP(wt_y, NFD, Wwt2, bwt2, wt_out, BNS, 128, NFD);
  k_zwhat<<<EWB((size_t)BNS * ZWH), 256, 0, stream>>>(wt_out, eps_what, z_what);

  // 25-27. depth head -> z_depth (K padded 129 -> 132 on both operands)
  k_depinp<<<EWB((size_t)BNS * DEPPAD), 256, 0, stream>>>(transit, z_what,
                                                          z_depth_pre, dep_inp);
  k_pad_w<<<EWB((size_t)DEPPAD * NFD), 256, 0, stream>>>(Wd1, Wd1p, DEPIN, DEPPAD, NFD);
  gemmP(dep_inp, DEPPAD, Wd1p, bd1, dep_y, BNS, NFD, DEPPAD);
  bnrun(dep_y, gdb, bdb, BNS, NFD);
  linear_small<<<EWB((size_t)BNS * 2), 256, 0, stream>>>(dep_y, Wd2, bd2, dep_out,
                                                         BNS, 2, NFD);
  k_zdepth<<<EWB(BNS), 256, 0, stream>>>(dep_out, eps_depth, z_depth);

  // 28-29. decoder
  gemmP(z_what, ZWH, Wdc1, bdc1, dec_h, BNS, 256, ZWH);
  k_celu_inplace<<<EWB((size_t)BNS * 256), 256, 0, stream>>>(dec_h, BNS * 256);
  gemmP(dec_h, 256, Wdc2, bdc2, dec_out, BNS, 4096, 256);

  // 30-31. inverse spatial transform (paste) -> d_out
  k_prepatt<<<EWB((size_t)BNS * GSD * GSD), 256, 0, stream>>>(dec_out, z_depth,
                                                              y_att, imp);
  k_inverse<<<EWB((size_t)BNS * IMGD * IMGD), 256, 0, stream>>>(y_att, imp, z_where,
                                                                out_y, out_imp);

  // 32. kl_z_what -> d_out
  k_kl_what<<<BSZ, 256, 0, stream>>>(enc_out, wt_out, z_pres_pre, out_klw);
}